// QuantizedBKCore_88673894793300
// MI455X (gfx1250) — compile-verified
//
#include <hip/hip_runtime.h>
#include <hip/hip_bf16.h>

#ifndef __has_builtin
#define __has_builtin(x) 0
#endif

// CDNA5 async global<->LDS DMA (ASYNCcnt-tracked). Guarded so the host pass
// (and toolchains lacking the builtins) fall back to plain coalesced loads.
#if defined(__HIP_DEVICE_COMPILE__) && \
    __has_builtin(__builtin_amdgcn_global_load_async_to_lds_b128) && \
    __has_builtin(__builtin_amdgcn_global_store_async_from_lds_b64)
#define BK_USE_ASYNC 1
#else
#define BK_USE_ASYNC 0
#endif

namespace {
constexpr int TPB  = 128;          // 4 wave32s per block
constexpr int K    = 32;           // elements owned per thread
constexpr int SEG  = TPB * K;      // 4096 elements per block (per row-segment)
constexpr int HALO = 64;           // warm-up; contraction <= 2^-1/step worst case
}

#if BK_USE_ASYNC
// Builtin signatures (clang-verified): data-sized int-vector pointers,
// global (addrspace 1) first, LDS (addrspace 3) second, imm offset, imm cpol.
typedef __attribute__((__vector_size__(16))) int bk_b128;
typedef __attribute__((__vector_size__(8)))  int bk_b64;
#define BK_AS_GLOBAL __attribute__((address_space(1)))
#define BK_AS_LDS    __attribute__((address_space(3)))

__device__ __forceinline__ void bk_wait_async() {
#if __has_builtin(__builtin_amdgcn_s_wait_asynccnt)
  __builtin_amdgcn_s_wait_asynccnt(0);
#else
  asm volatile("s_wait_asynccnt 0" ::: "memory");
#endif
}
#endif

// Fast reciprocal: denominators are >= 1 here (Im(z-a-L-R) >= 1) and the
// final results are rounded to integers, so 1-ulp v_rcp_f32 is plenty and
// shortens the serial dependence chain vs an IEEE divide.
__device__ __forceinline__ float bk_rcp(float x) {
#if defined(__HIP_DEVICE_COMPILE__) && __has_builtin(__builtin_amdgcn_rcpf)
  return __builtin_amdgcn_rcpf(x);
#else
  return 1.f / x;
#endif
}

// a_i = clip(hd_i + clip(rint(v_i), -128, 127), -10, 10)  (fake-quantized potential)
__device__ __forceinline__ float bk_a(float vv, float hdv) {
  float q = rintf(vv);
  q = fminf(127.f, fmaxf(-128.f, q));
  return fminf(10.f, fmaxf(-10.f, hdv + q));
}

__global__ __launch_bounds__(TPB) void bk_resolvent_fq_kernel(
    const float* __restrict__ v,     // (B, N)
    const float* __restrict__ hd,    // (N,)
    const float* __restrict__ hsub,  // (N-1,)
    const float* __restrict__ hsup,  // (N-1,)
    float*       __restrict__ out,   // (B, N, 2)
    int N)
{
  __shared__ __align__(16) float  vbuf[SEG + 2 * HALO];  // staged v window
  __shared__ __align__(16) float2 wbuf[SEG];             // W_i then quantized G_i

  const int seg = blockIdx.x;        // 2-D grid: no integer-divide prologue
  const int row = blockIdx.y;

  const int s0       = seg * SEG;
  const int segElems = min(SEG, N - s0);
  const int lo       = max(0, s0 - HALO);
  const int hi       = min(N, s0 + SEG + HALO);
  const int cnt      = hi - lo;
  const float* vrow  = v + (long long)row * N;
  const int t        = threadIdx.x;

  // -------- stage v[lo, hi) into LDS (coalesced async DMA) --------
  __builtin_prefetch(vrow + lo, 0, 1);
#if BK_USE_ASYNC
  if (((lo & 3) == 0) && ((cnt & 3) == 0)) {
    const int n4 = cnt >> 2;
    for (int m = t; m < n4; m += TPB) {
      __builtin_amdgcn_global_load_async_to_lds_b128(
          (BK_AS_GLOBAL bk_b128*)(const void*)(vrow + lo + 4 * m),
          (BK_AS_LDS    bk_b128*)(void*)&vbuf[4 * m], 0, 0);
    }
    bk_wait_async();
  } else
#endif
  {
    for (int m = t; m < cnt; m += TPB) vbuf[m] = vrow[lo + m];
  }
  __syncthreads();

  const int cs = s0 + t * K;                 // this thread's chunk [cs, ce)
  if (cs < N) {
    const int ce = min(cs + K, N);

    // ======== backward sweep: W_j = z - a_j - R_j  (z = i) ========
    // R_{j-1} = c_{j-1} * rcp(z - a_j - R_j); halo start R=0 (exact at N-1).
    const int j0 = min(N - 1, ce - 1 + HALO);
    float Rr = 0.f, Ri = 0.f;
    // halo warm-up: j in [ce, j0] (j >= ce >= 1, so hsub[j-1] is in range)
#pragma unroll 4
    for (int j = j0; j >= ce; --j) {
      const float aj = bk_a(vbuf[j - lo], hd[j]);
      const float wr = -aj - Rr;
      const float wi = 1.f - Ri;               // Im >= 1 always
      const float c  = hsub[j - 1] * hsup[j - 1];
      const float d  = c * bk_rcp(wr * wr + wi * wi);
      Rr = wr * d;
      Ri = -wi * d;
    }
    // main: store W_j for j in (cs, ce), then peeled j == cs (store only)
#pragma unroll 4
    for (int j = ce - 1; j > cs; --j) {
      const float aj = bk_a(vbuf[j - lo], hd[j]);
      const float wr = -aj - Rr;
      const float wi = 1.f - Ri;
      wbuf[j - s0] = make_float2(wr, wi);
      const float c = hsub[j - 1] * hsup[j - 1];
      const float d = c * bk_rcp(wr * wr + wi * wi);
      Rr = wr * d;
      Ri = -wi * d;
    }
    {
      const float aj = bk_a(vbuf[cs - lo], hd[cs]);
      wbuf[cs - s0] = make_float2(-aj - Rr, 1.f - Ri);
    }

    // ======== forward sweep + combine: G_i = 1/(W_i - L_i) ========
    // L_i = c_{i-1} * rcp(z - a_{i-1} - L_{i-1}); halo start L=0 (exact at 0).
    const int j1 = max(0, cs - HALO);
    float Lr = 0.f, Li = 0.f;
    float ap = bk_a(vbuf[j1 - lo], hd[j1]);
    // halo warm-up: i in (j1, cs) (i >= 1, so hsub[i-1] is in range)
#pragma unroll 4
    for (int i = j1 + 1; i < cs; ++i) {
      const float c  = hsub[i - 1] * hsup[i - 1];
      const float wr = -ap - Lr;
      const float wi = 1.f - Li;
      const float d  = c * bk_rcp(wr * wr + wi * wi);
      Lr = wr * d;
      Li = -wi * d;
      ap = bk_a(vbuf[i - lo], hd[i]);
    }
    // main: i in [cs, ce); the L-update is skipped only at i == j1 (cs == 0)
#pragma unroll 4
    for (int i = cs; i < ce; ++i) {
      if (i > j1) {
        const float c  = hsub[i - 1] * hsup[i - 1];
        const float wr = -ap - Lr;
        const float wi = 1.f - Li;
        const float d  = c * bk_rcp(wr * wr + wi * wi);
        Lr = wr * d;
        Li = -wi * d;
      }
      ap = bk_a(vbuf[i - lo], hd[i]);
      const float2 w  = wbuf[i - s0];
      const float  dr = w.x - Lr;
      const float  di = w.y - Li;
      const float  iv = bk_rcp(dr * dr + di * di);
      float Gr = dr * iv;
      float Gi = -di * iv;
      // clip to [-100,100] then fake-quantize (scale=1, zp=0) == rint
      Gr = rintf(fminf(100.f, fmaxf(-100.f, Gr)));
      Gi = rintf(fminf(100.f, fmaxf(-100.f, Gi)));
      wbuf[i - s0] = make_float2(Gr, Gi);       // re-use LDS as output tile
    }
  }
  __syncthreads();

  // -------- packed (re,im) writeout, coalesced from LDS --------
  float2* out2 = reinterpret_cast<float2*>(out) + ((long long)row * N + s0);
#if BK_USE_ASYNC
  for (int m = t; m < segElems; m += TPB) {
    __builtin_amdgcn_global_store_async_from_lds_b64(
        (BK_AS_GLOBAL bk_b64*)(void*)(out2 + m),
        (BK_AS_LDS    bk_b64*)(void*)&wbuf[m], 0, 0);
  }
  bk_wait_async();   // drain before LDS re-use / wave exit
#else
  for (int m = t; m < segElems; m += TPB) out2[m] = wbuf[m];
#endif
}

extern "C" void kernel_launch(void* const* d_in, const int* in_sizes, int n_in,
                              void* d_out, int out_size, void* d_ws, size_t ws_size,
                              hipStream_t stream) {
  const float* v    = (const float*)d_in[0];
  const float* hd   = (const float*)d_in[1];
  const float* hsub = (const float*)d_in[2];
  const float* hsup = (const float*)d_in[3];
  float* out = (float*)d_out;

  const int N    = in_sizes[1];            // h0_diag has N elements
  const int B    = in_sizes[0] / N;        // v is B*N
  const int nseg = (N + SEG - 1) / SEG;

  dim3 grid((unsigned)nseg, (unsigned)B), block(TPB);
  bk_resolvent_fq_kernel<<<grid, block, 0, stream>>>(v, hd, hsub, hsup, out, N);
}